// WarpFromDepth_23252952940824
// MI455X (gfx1250) — compile-verified
//
#include <hip/hip_runtime.h>

// Problem geometry (fixed by the reference).
#define NB 8
#define CC 3
#define HDm 1024
#define WDm 1280
#define HSm 800
#define WSm 1280

static constexpr int HWd = HDm * WDm;   // depth/out plane size (1,310,720)
static constexpr int HWs = HSm * WSm;   // src plane size (1,024,000)

typedef float f32x4 __attribute__((ext_vector_type(4)));

__global__ __launch_bounds__(256) void warp_from_depth_kernel(
    const float* __restrict__ depth,   // [N,1,HD,WD]
    const float* __restrict__ src,     // [N,C,HS,WS]
    const float* __restrict__ abc,     // [3,HD,WD]
    const float* __restrict__ tran,    // [3]
    const float* __restrict__ intr,    // [4]
    float*       __restrict__ out)     // [N,C,HD,WD]
{
    // 2D grid: blockIdx.y = batch (no integer division), blockIdx.x covers one
    // depth plane in quads of 4 pixels. WD%4==0 so a quad never crosses a row;
    // all float4 accesses are 16B aligned.
    const int n   = blockIdx.y;
    const int pix = (blockIdx.x * blockDim.x + threadIdx.x) * 4;
    if (pix >= HWd) return;

    // Uniform calibration scalars (scalarized to s_load by the compiler).
    const float tx = tran[0], ty = tran[1], tz = tran[2];
    const float fu = intr[0], fv = intr[1], du = intr[2], dv = intr[3];

    // depth is read exactly once -> non-temporal load (don't pollute L2;
    // keep L2 for the src planes the gathers below depend on).
    const f32x4 d4 = __builtin_nontemporal_load((const f32x4*)(depth + (size_t)n * HWd + pix));
    // abc is re-read every batch -> regular temporal policy (L2 resident).
    const f32x4 a4 = *(const f32x4*)(abc + pix);
    const f32x4 b4 = *(const f32x4*)(abc + HWd + pix);
    const f32x4 c4 = *(const f32x4*)(abc + 2 * HWd + pix);

    const float* __restrict__ pc0 = src + (size_t)n * CC * HWs; // channel 0 plane
    const float* __restrict__ pc1 = pc0 + HWs;
    const float* __restrict__ pc2 = pc1 + HWs;

    float r0[4], r1[4], r2[4];

    #pragma unroll
    for (int j = 0; j < 4; ++j) {
        const float d = d4[j];
        const float a = a4[j];
        const float b = b4[j];
        const float c = c4[j];

        // Projective transform into projector plane.
        const float denom = fmaf(c, d, tz);
        const float inv   = 1.0f / denom;
        const float xx    = fmaf(a, d, tx) * inv;
        const float yy    = fmaf(b, d, ty) * inv;
        const float uu    = fmaf(fu, xx, du);
        const float vv    = fmaf(fv, yy, dv);

        // Faithful normalize -> unnormalize round trip (matches reference).
        const float gx = 2.0f * uu * (1.0f / (float)(WSm - 1)) - 1.0f;
        const float gy = 2.0f * vv * (1.0f / (float)(HSm - 1)) - 1.0f;
        float x = ((gx + 1.0f) * (float)WSm - 1.0f) * 0.5f;
        float y = ((gy + 1.0f) * (float)HSm - 1.0f) * 0.5f;

        // Border clamp.
        x = fminf(fmaxf(x, 0.0f), (float)(WSm - 1));
        y = fminf(fmaxf(y, 0.0f), (float)(HSm - 1));

        const float x0f = floorf(x), y0f = floorf(y);
        const float wx = x - x0f,  wy = y - y0f;
        const int x0 = (int)x0f,   y0 = (int)y0f;
        const int x1 = min(x0 + 1, WSm - 1);
        const int y1 = min(y0 + 1, HSm - 1);

        const int o00 = y0 * WSm + x0;
        const int o01 = y0 * WSm + x1;
        const int o10 = y1 * WSm + x0;
        const int o11 = y1 * WSm + x1;

        const float w00 = (1.0f - wy) * (1.0f - wx);
        const float w01 = (1.0f - wy) * wx;
        const float w10 = wy * (1.0f - wx);
        const float w11 = wy * wx;

        // Issue all 12 gathers up front (independent addresses -> deep MLP,
        // mostly L2 hits since src planes stay resident).
        const float s0a = pc0[o00], s0b = pc0[o01], s0c = pc0[o10], s0d = pc0[o11];
        const float s1a = pc1[o00], s1b = pc1[o01], s1c = pc1[o10], s1d = pc1[o11];
        const float s2a = pc2[o00], s2b = pc2[o01], s2c = pc2[o10], s2d = pc2[o11];

        r0[j] = fmaf(s0a, w00, fmaf(s0b, w01, fmaf(s0c, w10, s0d * w11)));
        r1[j] = fmaf(s1a, w00, fmaf(s1b, w01, fmaf(s1c, w10, s1d * w11)));
        r2[j] = fmaf(s2a, w00, fmaf(s2b, w01, fmaf(s2c, w10, s2d * w11)));
    }

    // Output is write-once streaming -> non-temporal stores so the 126 MB of
    // writes do not evict the gathered src planes from the 192 MB L2.
    float* __restrict__ o = out + (size_t)n * CC * HWd + pix;
    f32x4 v0 = {r0[0], r0[1], r0[2], r0[3]};
    f32x4 v1 = {r1[0], r1[1], r1[2], r1[3]};
    f32x4 v2 = {r2[0], r2[1], r2[2], r2[3]};
    __builtin_nontemporal_store(v0, (f32x4*)(o));
    __builtin_nontemporal_store(v1, (f32x4*)(o + HWd));
    __builtin_nontemporal_store(v2, (f32x4*)(o + 2 * HWd));
}

extern "C" void kernel_launch(void* const* d_in, const int* in_sizes, int n_in,
                              void* d_out, int out_size, void* d_ws, size_t ws_size,
                              hipStream_t stream) {
    (void)in_sizes; (void)n_in; (void)out_size; (void)d_ws; (void)ws_size;
    const float* depth = (const float*)d_in[0];   // [N,1,HD,WD]
    const float* src   = (const float*)d_in[1];   // [N,C,HS,WS]
    const float* abc   = (const float*)d_in[2];   // [3,HD,WD]
    const float* tran  = (const float*)d_in[3];   // [3]
    const float* intr  = (const float*)d_in[4];   // [4]
    float*       out   = (float*)d_out;           // [N,C,HD,WD]

    const int quads_per_plane = HWd / 4;                     // 327,680
    const int block = 256;                                   // 8 wave32s
    dim3 grid((quads_per_plane + block - 1) / block, NB);    // (1280, 8)
    warp_from_depth_kernel<<<grid, dim3(block), 0, stream>>>(depth, src, abc, tran, intr, out);
}